// DCellOpt_74766790689034
// MI455X (gfx1250) — compile-verified
//
#include <hip/hip_runtime.h>

#define B_    128
#define T_    2048
#define S_    8
#define TPS_  256
#define G_    256
#define DOUT_ 64
#define C_    4
#define KC    64
#define LDA   68   // padded LDS row stride in floats (16B-aligned rows, conflict-free frags)

typedef float v2f __attribute__((ext_vector_type(2)));
typedef float v8f __attribute__((ext_vector_type(8)));

#if __has_builtin(__builtin_amdgcn_global_load_async_to_lds_b128)
#define HAVE_ASYNC 1
#define NBUF 2
#define BUFIDX(k) ((k) & 1)
typedef int afi4 __attribute__((vector_size(16)));           // matches builtin's V4i
typedef __attribute__((address_space(1))) afi4* g_v4i_ptr;   // global (AS1) v4i*
typedef __attribute__((address_space(3))) afi4* l_v4i_ptr;   // LDS (AS3) v4i*
__device__ __forceinline__ void async_cp16(void* lds, const void* g) {
    __builtin_amdgcn_global_load_async_to_lds_b128(
        (g_v4i_ptr)g, (l_v4i_ptr)lds, /*offset=*/0, /*cpol=*/0);
}
__device__ __forceinline__ void wait_async0() {
#if __has_builtin(__builtin_amdgcn_s_wait_asynccnt)
    __builtin_amdgcn_s_wait_asynccnt(0);
#else
    asm volatile("s_wait_asynccnt 0x0" ::: "memory");
#endif
}
#else
#define NBUF 1
#define BUFIDX(k) 0
#endif

__launch_bounds__(256, 1)
__global__ void dcell_stratum_kernel(
    const float* __restrict__ gene,    // (B,T,G)
    const float* __restrict__ Wmat,    // leaf: (TPS,256,64); internal: (1792,512,64)
    const float* __restrict__ gamma,   // (T,64)
    const float* __restrict__ beta,    // (T,64)
    const float* __restrict__ Wh,      // (T,64,1)
    const float* __restrict__ bh,      // (T,1)
    const int*   __restrict__ ci,      // (T,4)
    const float* __restrict__ prevH,   // (B,TPS,64)
    float*       __restrict__ curH,    // (B,TPS,64)
    float*       __restrict__ scores,  // (B,T)
    int s, int din, int is_leaf, int write_h)
{
    __shared__ float sA[NBUF][128 * LDA];   // A staging (double-buffered if async); sA[0] reused as Z
    __shared__ float sW[DOUT_ * LDA];       // W^T staging: sW[n][k]
    __shared__ float sScale[DOUT_];
    __shared__ float sShift[DOUT_];

    const int tl   = blockIdx.x;
    const int t    = s * TPS_ + tl;
    const int tid  = threadIdx.x;
    const int lane = tid & 31;
    const int wave = tid >> 5;
    const int half = lane >> 4;
    const int n0   = lane & 15;
    const int rowA = wave * 16 + n0;

    const int nchunks   = din / KC;         // 4 (leaf) or 8 (internal)
    const int childBase = (s + 1) * TPS_;
    const int widx      = is_leaf ? tl : t;

    // per-thread staging coordinates
    const int ar  = tid >> 1;               // A row 0..127
    const int akq = (tid & 1) * 32;         // A col offset 0/32
    const int wn4 = (tid & 15) * 4;         // W cols n4..n4+3
    const int wk0 = tid >> 4;               // W base k 0..15

    // global source for this thread's 32-float A slice of chunk kc (nullptr => zeros)
    auto a_src = [&](int kc) -> const float* {
        if (!is_leaf && kc < C_) {
            int tc = ci[t * C_ + kc];                       // uniform across block
            if (tc < 0) return (const float*)0;
            return prevH + ((size_t)ar * TPS_ + (tc - childBase)) * DOUT_ + akq;
        }
        int goff = is_leaf ? kc * KC : (kc - C_) * KC;
        return gene + ((size_t)ar * T_ + t) * G_ + goff + akq;
    };

#ifndef HAVE_ASYNC
    float4 aReg[8];
#endif
    float4 wReg[4];

    auto a_issue = [&](int kc, int buf) {
        const float* src = a_src(kc);
#ifdef HAVE_ASYNC
        float* dst = &sA[buf][ar * LDA + akq];
        if (src) {
            #pragma unroll
            for (int i = 0; i < 8; ++i) async_cp16(dst + 4 * i, src + 4 * i);
        }
#else
        if (src) {
            #pragma unroll
            for (int i = 0; i < 8; ++i) aReg[i] = *(const float4*)(src + 4 * i);
        } else {
            float4 z4 = make_float4(0.f, 0.f, 0.f, 0.f);
            #pragma unroll
            for (int i = 0; i < 8; ++i) aReg[i] = z4;
        }
#endif
    };
    auto a_commit = [&](int kc, int buf) {
        float* dst = &sA[buf][ar * LDA + akq];
#ifdef HAVE_ASYNC
        if (!a_src(kc)) {                                   // zero-fill (uniform branch)
            float4 z4 = make_float4(0.f, 0.f, 0.f, 0.f);
            #pragma unroll
            for (int i = 0; i < 8; ++i) *(float4*)(dst + 4 * i) = z4;
        }
        wait_async0();
#else
        #pragma unroll
        for (int i = 0; i < 8; ++i) *(float4*)(dst + 4 * i) = aReg[i];
#endif
    };
    auto w_issue = [&](int kc) {
        const float* wsrc = Wmat + ((size_t)widx * din + (size_t)kc * KC) * DOUT_;
        #pragma unroll
        for (int i = 0; i < 4; ++i)
            wReg[i] = *(const float4*)&wsrc[(wk0 + 16 * i) * DOUT_ + wn4];
    };
    auto w_commit = [&]() {
        #pragma unroll
        for (int i = 0; i < 4; ++i) {
            int kk = wk0 + 16 * i;
            sW[(wn4 + 0) * LDA + kk] = wReg[i].x;
            sW[(wn4 + 1) * LDA + kk] = wReg[i].y;
            sW[(wn4 + 2) * LDA + kk] = wReg[i].z;
            sW[(wn4 + 3) * LDA + kk] = wReg[i].w;
        }
    };

    v8f acc0 = {}, acc1 = {}, acc2 = {}, acc3 = {};

    // ---- prologue: stage chunk 0 ----
    w_issue(0);
    a_issue(0, 0);
    w_commit();
    a_commit(0, 0);
    __syncthreads();

    for (int kc = 0; kc < nchunks; ++kc) {
        const int  cb      = BUFIDX(kc);
        const int  nb      = BUFIDX(kc + 1);
        const bool hasNext = (kc + 1) < nchunks;

        // overlap: start fetching chunk kc+1 while we do chunk kc's math
        if (hasNext) { a_issue(kc + 1, nb); w_issue(kc + 1); }

        const float* Ab = &sA[cb][0];
        #pragma unroll
        for (int ks = 0; ks < KC; ks += 4) {
            v2f a  = *(const v2f*)&Ab[rowA * LDA + ks + 2 * half];
            v2f b0 = *(const v2f*)&sW[( 0 + n0) * LDA + ks + 2 * half];
            v2f b1 = *(const v2f*)&sW[(16 + n0) * LDA + ks + 2 * half];
            v2f b2 = *(const v2f*)&sW[(32 + n0) * LDA + ks + 2 * half];
            v2f b3 = *(const v2f*)&sW[(48 + n0) * LDA + ks + 2 * half];
            acc0 = __builtin_amdgcn_wmma_f32_16x16x4_f32(false, a, false, b0, (short)0, acc0, false, false);
            acc1 = __builtin_amdgcn_wmma_f32_16x16x4_f32(false, a, false, b1, (short)0, acc1, false, false);
            acc2 = __builtin_amdgcn_wmma_f32_16x16x4_f32(false, a, false, b2, (short)0, acc2, false, false);
            acc3 = __builtin_amdgcn_wmma_f32_16x16x4_f32(false, a, false, b3, (short)0, acc3, false, false);
        }
        __syncthreads();                      // all waves done with sW and sA[cb]
        if (hasNext) { w_commit(); a_commit(kc + 1, nb); }
        __syncthreads();                      // chunk kc+1 published
    }

    // ---- spill z tiles to LDS (C/D layout: VGPR j -> M = j + 8*half, N = n0) ----
    {
        float* Z = &sA[0][0];
        const int mrow = wave * 16 + 8 * half;
        #pragma unroll
        for (int j = 0; j < 8; ++j) {
            Z[(mrow + j) * LDA +  0 + n0] = acc0[j];
            Z[(mrow + j) * LDA + 16 + n0] = acc1[j];
            Z[(mrow + j) * LDA + 32 + n0] = acc2[j];
            Z[(mrow + j) * LDA + 48 + n0] = acc3[j];
        }
    }
    __syncthreads();

    // ---- batch-norm statistics per output channel (bias cancels in z - mu) ----
    if (tid < DOUT_) {
        float sum = 0.f, ss = 0.f;
        for (int r = 0; r < B_; ++r) {
            float v = sA[0][r * LDA + tid];
            sum += v; ss += v * v;
        }
        float mu  = sum * (1.0f / B_);
        float var = ss * (1.0f / B_) - mu * mu;   // biased var, matches jnp.var
        float rs  = rsqrtf(var + 1e-5f);
        float sc  = gamma[(size_t)t * DOUT_ + tid] * rs;
        sScale[tid] = sc;
        sShift[tid] = beta[(size_t)t * DOUT_ + tid] - mu * sc;
    }
    __syncthreads();

    // ---- tanh, h writeback, score head ----
    {
        const int r  = tid >> 1;
        const int nq = (tid & 1) * 32;
        const float* whp = Wh + (size_t)t * DOUT_;
        float p = 0.f;
        #pragma unroll 8
        for (int i = 0; i < 32; ++i) {
            int n = nq + i;
            float z = sA[0][r * LDA + n];
            float h = tanhf(fmaf(z, sScale[n], sShift[n]));
            if (write_h)
                curH[((size_t)r * TPS_ + tl) * DOUT_ + n] = h;
            p = fmaf(h, whp[n], p);
        }
        p += __shfl_xor(p, 1, 32);
        if ((tid & 1) == 0)
            scores[(size_t)r * T_ + t] = p + bh[t];
    }
}

extern "C" void kernel_launch(void* const* d_in, const int* in_sizes, int n_in,
                              void* d_out, int out_size, void* d_ws, size_t ws_size,
                              hipStream_t stream) {
    const float* gene   = (const float*)d_in[0];
    const float* W_leaf = (const float*)d_in[1];
    // d_in[2] = b_leaf, d_in[4] = b_int: mathematically cancelled by batch-norm
    const float* W_int  = (const float*)d_in[3];
    const float* gamma  = (const float*)d_in[5];
    const float* beta   = (const float*)d_in[6];
    const float* Wh     = (const float*)d_in[7];
    const float* bh     = (const float*)d_in[8];
    const int*   ci     = (const int*)d_in[9];
    float* scores = (float*)d_out;

    const size_t bufElems = (size_t)B_ * TPS_ * DOUT_;   // 8 MB each
    if (ws_size < 2 * bufElems * sizeof(float)) return;  // need 16 MB ping-pong
    float* buf0 = (float*)d_ws;
    float* buf1 = buf0 + bufElems;

    for (int s = S_ - 1; s >= 0; --s) {
        float*       cur  = (s & 1) ? buf1 : buf0;
        const float* prev = (s & 1) ? buf0 : buf1;
        const int leaf = (s == S_ - 1) ? 1 : 0;
        dcell_stratum_kernel<<<TPS_, 256, 0, stream>>>(
            gene, leaf ? W_leaf : W_int, gamma, beta, Wh, bh, ci,
            prev, cur, scores,
            s, leaf ? G_ : (C_ * DOUT_ + G_), leaf, (s > 0) ? 1 : 0);
    }
}